// S4DKernel_24292335026633
// MI455X (gfx1250) — compile-verified
//
#include <hip/hip_runtime.h>
#include <math.h>

// S4D convolution-kernel materialization for MI455X (gfx1250).
// K[h,l] = 2*Re( sum_n Cc[h,n] * dtA[h,n]^l ), H=512, N2=32, L=8192.
// Per-head f32 GEMM (512x64)x(64x16) on V_WMMA_F32_16X16X4_F32, with all
// transcendentals hoisted to setup; A-fragments advance by a complex-FMA
// recurrence P *= dtA^1024 between tiles.

constexpr int kH  = 512;
constexpr int kN2 = 32;
constexpr int kL  = 8192;

typedef float v2f __attribute__((ext_vector_type(2)));
typedef float v8f __attribute__((ext_vector_type(8)));

// ---------------------------------------------------------------------------
// Pass 1: bilinear discretization + complex log, one thread per (h, n).
// ws[(h*N2+n)] = float4{ log|dtA|, arg(dtA), Re(Cc), Im(Cc) }
// ---------------------------------------------------------------------------
__global__ __launch_bounds__(256) void s4d_precompute(
    const float* __restrict__ log_dt, const float* __restrict__ B_real,
    const float* __restrict__ B_imag, const float* __restrict__ C,
    const float* __restrict__ A_real, const float* __restrict__ A_imag,
    float4* __restrict__ ws)
{
    int i = blockIdx.x * blockDim.x + threadIdx.x;
    if (i >= kH * kN2) return;
    int h = i / kN2;

    float dt  = __expf(log_dt[h]);
    float are = -__expf(A_real[i]) - 1e-6f;
    float aim = A_imag[i];

    // denom = 1 - dt*A/2 + 1e-6 ; num = 1 + dt*A/2
    float dre = 1.0f - 0.5f * dt * are + 1e-6f;
    float dim = -0.5f * dt * aim;
    float nre = 1.0f + 0.5f * dt * are;
    float nim = 0.5f * dt * aim;
    float inv = 1.0f / (dre * dre + dim * dim);

    // dtA = num / denom
    float tre = (nre * dre + nim * dim) * inv;
    float tim = (nim * dre - nre * dim) * inv;

    // dtB = dt * B / denom ; Cc = C * dtB   (C is real)
    float br = B_real[i], bi = B_imag[i];
    float cv = C[i];
    float ccre = cv * dt * (br * dre + bi * dim) * inv;
    float ccim = cv * dt * (bi * dre - br * dim) * inv;

    // complex log of dtA
    float lre = 0.5f * __logf(tre * tre + tim * tim);
    float lim = atan2f(tim, tre);

    ws[i] = make_float4(lre, lim, ccre, ccim);
}

// ---------------------------------------------------------------------------
// Pass 2: per-head GEMM on WMMA. Block = 1 head, 4 waves; wave w covers
// M-tiles {w, w+4, ..., w+28} (16 t-rows each), chaining 16 K=4 WMMAs over
// the 64-wide (Re/Im x 32 modes) contraction.
//
// Fragment mapping (wave32, ISA VGPR layouts):
//   A 16x4 f32 : lane%16 = M-row, lane/16 selects K in {0,1}/{2,3}
//                -> lane owns complex P[n], n = 2*kk + half, a = {Re P, Im P}
//   B 4x16 f32 : lane%16 = N-col j, same half split
//                -> b = {Re Q, -Im Q},  Q = dtA^j
//   D 16x16 f32: VGPR r -> row (r + 8*half), col lane%16
// ---------------------------------------------------------------------------
__global__ __launch_bounds__(128) void s4d_wmma(const float4* __restrict__ ws,
                                                float* __restrict__ out)
{
    const int h    = blockIdx.x;
    const int lane = threadIdx.x & 31;
    const int wave = threadIdx.x >> 5;   // 0..3
    const int m    = lane & 15;          // M-row of A frag / N-col of B,D frags
    const int half = lane >> 4;          // 0 -> even modes, 1 -> odd modes

    // Per-lane state for its 16 modes (n = 2*kk + half):
    //   q  = dtA^j (j = lane%16), imag part pre-negated for the B frag
    //   p  = Cc * dtA^(16*(16*wave + m))           (A-frag starts, tile 0)
    //   st = dtA^1024                              (per-tile advance step)
    float qre[16], qim[16], pre[16], pim[16], sre[16], sim[16];
    const float4* p4 = ws + h * kN2;
    const float jj = (float)m;
    const float s0 = 16.0f * (float)(wave * 16 + m);
#pragma unroll
    for (int kk = 0; kk < 16; ++kk) {
        float4 v = p4[2 * kk + half];
        float e, sn, cs;
        // Q = dtA^j  (store -Im so the B frag is a straight copy)
        e = __expf(v.x * jj);
        __sincosf(v.y * jj, &sn, &cs);
        qre[kk] = e * cs;
        qim[kk] = -e * sn;
        // P0 = Cc * dtA^s0
        e = __expf(v.x * s0);
        __sincosf(v.y * s0, &sn, &cs);
        float wr = e * cs, wi = e * sn;
        pre[kk] = v.z * wr - v.w * wi;
        pim[kk] = v.z * wi + v.w * wr;
        // step = dtA^1024
        e = __expf(v.x * 1024.0f);
        __sincosf(v.y * 1024.0f, &sn, &cs);
        sre[kk] = e * cs;
        sim[kk] = e * sn;
    }

    float* __restrict__ op = out + (size_t)h * kL;

    for (int it = 0; it < 8; ++it) {
        const int t_base = (wave + 4 * it) * 16;      // first t-row of tile
        v8f acc = {};
#pragma unroll
        for (int kk = 0; kk < 16; ++kk) {
            v2f a; a.x = pre[kk]; a.y = pim[kk];
            v2f b; b.x = qre[kk]; b.y = qim[kk];
            acc = __builtin_amdgcn_wmma_f32_16x16x4_f32(
                false, a, false, b, (short)0, acc, false, false);
            // Advance P for the next tile: P *= dtA^1024 (complex FMA).
            float nr = pre[kk] * sre[kk] - pim[kk] * sim[kk];
            float ni = pre[kk] * sim[kk] + pim[kk] * sre[kk];
            pre[kk] = nr;
            pim[kk] = ni;
        }
        // D layout: VGPR r -> row (r + 8*half), col m.  l = 16*t + j.
#pragma unroll
        for (int r = 0; r < 8; ++r) {
            int trow = t_base + r + 8 * half;
            op[trow * 16 + m] = 2.0f * acc[r];
        }
    }
}

// ---------------------------------------------------------------------------
extern "C" void kernel_launch(void* const* d_in, const int* in_sizes, int n_in,
                              void* d_out, int out_size, void* d_ws, size_t ws_size,
                              hipStream_t stream) {
    (void)in_sizes; (void)n_in; (void)out_size; (void)ws_size;
    const float* log_dt = (const float*)d_in[0];
    const float* B_real = (const float*)d_in[1];
    const float* B_imag = (const float*)d_in[2];
    const float* C      = (const float*)d_in[3];
    // d_in[4] = Kc (unused by the reference math)
    const float* A_real = (const float*)d_in[5];
    const float* A_imag = (const float*)d_in[6];
    // d_in[7] = L scalar (== 8192, baked in as kL)

    float4* ws  = (float4*)d_ws;
    float*  out = (float*)d_out;

    s4d_precompute<<<(kH * kN2 + 255) / 256, 256, 0, stream>>>(
        log_dt, B_real, B_imag, C, A_real, A_imag, ws);
    s4d_wmma<<<kH, 128, 0, stream>>>(ws, out);
}